// PatchBasedEmbedding_35141422416305
// MI455X (gfx1250) — compile-verified
//
#include <hip/hip_runtime.h>

typedef float v2f __attribute__((ext_vector_type(2)));
typedef float v8f __attribute__((ext_vector_type(8)));

#define F2      4096          // filters incl. flipped copy
#define PITCH   17            // LDS pitch (gcd(17,64)=1 -> conflict-free column reads)
#define KDIM    108           // 3*6*6
#define NPIX    16            // pixels per workgroup tile
#define NPIX_TOT 11664        // 16*27*27  (== 729 * 16 exactly)
#define SMEM_BYTES 301952

// monotone key: ascending float order == ascending unsigned key order
__device__ __forceinline__ unsigned mkey(float f) {
  unsigned u = __float_as_uint(f);
  return u ^ ((u >> 31) ? 0xFFFFFFFFu : 0x80000000u);
}

// Rearrange conv weights (rows 0..2047 = kernel, 2048..4095 = W-flipped kernel)
// into WMMA A-fragment order: [mtile 256][kk 27][lane 32][2 floats], so the GEMM
// loads one coalesced float2 per lane per fragment.
// A 16x4 f32 layout: lanes 0-15 hold M=0..15 / K={kk*4+0, kk*4+1},
//                    lanes 16-31 hold M=0..15 / K={kk*4+2, kk*4+3}.
__global__ void prep_weights(const float* __restrict__ kern,
                             float* __restrict__ aprep) {
  int idx = blockIdx.x * 256 + threadIdx.x;
  if (idx >= 256 * 27 * 64) return;
  int j    = idx & 1;
  int lane = (idx >> 1) & 31;
  int kk   = (idx >> 6) % 27;
  int mt   = idx / (27 * 64);
  int M = mt * 16 + (lane & 15);
  int k = kk * 4 + ((lane >> 4) << 1) + j;     // 0..107 always (27*4 == 108)
  float v;
  if (M < 2048) {
    v = kern[M * KDIM + k];
  } else {
    int kw = k % 6;                            // flip along last axis
    v = kern[(M - 2048) * KDIM + (k - kw + (5 - kw))];
  }
  aprep[idx] = v;
}

__global__ __launch_bounds__(256)
void conv_select_kernel(const float* __restrict__ images,
                        const float* __restrict__ bias,
                        const float* __restrict__ aprep,
                        float* __restrict__ out) {
  extern __shared__ float smem[];
  float*    convv  = smem;                        // [4096][17]   272 KB
  float*    bias_s = convv + F2 * PITCH;          // [2048]
  float*    Bp     = bias_s + 2048;               // [108][16] im2col tile
  unsigned* hist   = (unsigned*)(Bp + KDIM * NPIX); // [8 waves][256]
  unsigned* wsel   = hist + 8 * 256;              // [8 waves][2] = {rank, prefix}
  unsigned* kthk   = wsel + 16;                   // [16] kth key per pixel

  const int tid  = threadIdx.x;
  const int lane = tid & 31;
  const int wv   = tid >> 5;                      // 8 wave32's
  const int g    = blockIdx.x;                    // 729 blocks * 16 pixels

  // ---- stage bias + im2col patch tile into LDS ----
  for (int i = tid; i < 2048; i += 256) bias_s[i] = bias[i];
  for (int e = tid; e < KDIM * NPIX; e += 256) {
    int k = e >> 4;
    int p = e & 15;
    int pix = g * NPIX + p;
    int n = pix / 729, q = pix % 729;
    int oh = q / 27, ow = q % 27;
    int c = k / 36, r = (k % 36) / 6, kw = k % 6;
    Bp[k * NPIX + p] = images[((n * 3 + c) * 32 + (oh + r)) * 32 + (ow + kw)];
  }
  __syncthreads();

  // ---- B fragments live in VGPRs for the whole GEMM (27 x float2 = 54 VGPRs)
  // B 4x16 layout: v0 = B[k0+0][col], v1 = B[k0+1][col]; k0 = (lane<16 ? 0 : 2)
  v2f bfrag[27];
  {
    int k0  = (lane >> 4) << 1;
    int col = lane & 15;
#pragma unroll
    for (int kk = 0; kk < 27; ++kk) {
      bfrag[kk].x = Bp[(kk * 4 + k0 + 0) * NPIX + col];
      bfrag[kk].y = Bp[(kk * 4 + k0 + 1) * NPIX + col];
    }
  }

  // ---- GEMM: 256 M-tiles of 16 filters; wave wv owns tiles wv, wv+8, ... ----
  const v2f* Ap = (const v2f*)aprep;
  for (int t = 0; t < 32; ++t) {
    int mt = wv + t * 8;
    v8f acc = {};
    const v2f* a = Ap + (mt * 27) * 32 + lane;
#pragma unroll
    for (int kk = 0; kk < 27; ++kk) {
      v2f af = a[kk * 32];
      acc = __builtin_amdgcn_wmma_f32_16x16x4_f32(
          false, af, false, bfrag[kk], (short)0, acc, false, false);
    }
    // D layout: VGPR r -> row (r + 8*half), col = lane&15
    int col   = lane & 15;
    int rbase = mt * 16 + ((lane >> 4) << 3);
#pragma unroll
    for (int r = 0; r < 8; ++r)
      convv[(rbase + r) * PITCH + col] = acc[r];
  }
  __syncthreads();

  // ---- exact radix-select of the 129th-smallest of 8192 per pixel ----
  // 8 waves x 2 pixels each; 4 passes of 8-bit histograms, MSB first.
  for (int pi = 0; pi < 2; ++pi) {
    int p = wv * 2 + pi;
    if (lane == 0) { wsel[wv * 2] = 128u; wsel[wv * 2 + 1] = 0u; }
    for (int pass = 0; pass < 4; ++pass) {
      int shift = 24 - pass * 8;
#pragma unroll
      for (int h = 0; h < 8; ++h) hist[wv * 256 + lane * 8 + h] = 0u;
      __syncthreads();
      unsigned pfx    = wsel[wv * 2 + 1];
      unsigned himask = pass ? (0xFFFFFFFFu << (shift + 8)) : 0u;
      for (int it = 0; it < 128; ++it) {
        int jv   = it * 32 + lane;
        float v  = convv[jv * PITCH + p];
        float bb = bias_s[jv & 2047];
#pragma unroll
        for (int s = 0; s < 2; ++s) {
          float d = s ? (v + bb) : (bb - v);       // {+conv+b, -conv+b}
          unsigned key = mkey(d);
          if (((key ^ pfx) & himask) == 0)
            atomicAdd(&hist[wv * 256 + ((key >> shift) & 255)], 1u);
        }
      }
      __syncthreads();
      if (lane == 0) {
        unsigned rank = wsel[wv * 2];
        unsigned pfx2 = wsel[wv * 2 + 1];
        unsigned cum = 0;
        for (int bin = 0; bin < 256; ++bin) {
          unsigned cnt = hist[wv * 256 + bin];
          if (cum + cnt > rank) {
            wsel[wv * 2]     = rank - cum;
            wsel[wv * 2 + 1] = pfx2 | ((unsigned)bin << shift);
            break;
          }
          cum += cnt;
        }
      }
      __syncthreads();
    }
    if (lane == 0) kthk[p] = wsel[wv * 2 + 1];     // full 32-bit key of kth value
    __syncthreads();
  }

  // ---- mask + coalesced store: lanes 0..15 of each 16-thread group cover
  // the 16 consecutive pixels of one channel -> 64B contiguous runs.
  {
    int col  = tid & 15;
    int crow = tid >> 4;
    int pix = g * NPIX + col;
    int n = pix / 729, q = pix % 729;
    unsigned kq = kthk[col];
    long base = (long)n * 8192 * 729 + q;
    for (int c0 = 0; c0 < 8192; c0 += 16) {
      int c  = c0 + crow;
      int jv = c & 4095;                 // 0..2047 conv, 2048..4095 conv_flip
      float v  = convv[jv * PITCH + col];
      float bb = bias_s[c & 2047];
      float d  = (c < 4096) ? (bb - v) : (v + bb);
      out[base + (long)c * 729] = (mkey(d) < kq) ? 1.0f : 0.0f;
    }
  }
}

extern "C" void kernel_launch(void* const* d_in, const int* in_sizes, int n_in,
                              void* d_out, int out_size, void* d_ws, size_t ws_size,
                              hipStream_t stream) {
  const float* images = (const float*)d_in[0];   // (16,3,32,32)
  const float* kern   = (const float*)d_in[1];   // (2048,3,6,6)
  const float* bias   = (const float*)d_in[2];   // (2048,)
  float* aprep = (float*)d_ws;                   // 442368 floats = 1.77 MB

  prep_weights<<<(256 * 27 * 64 + 255) / 256, 256, 0, stream>>>(kern, aprep);
  conv_select_kernel<<<NPIX_TOT / NPIX, 256, SMEM_BYTES, stream>>>(
      images, bias, aprep, (float*)d_out);
}